// MHA_52080773431701
// MI455X (gfx1250) — compile-verified
//
#include <hip/hip_runtime.h>

// MHA with diagonal-softmax simplification, exact fp32 via V_WMMA_F32_16X16X4_F32.
// Workspace: Q,K,V (3x33.5MB) + O (33.5MB) + WoT (4MB) + PqT/PkT/PvT (12MB) ~ 152MB.

typedef float v2f __attribute__((ext_vector_type(2)));
typedef float v8f __attribute__((ext_vector_type(8)));

#define B_  8
#define M_  1024
#define D_  1024
#define H_  16
#define DK_ 64

__device__ __forceinline__ v8f wmma4f32(v2f a, v2f b, v8f c) {
  return __builtin_amdgcn_wmma_f32_16x16x4_f32(false, a, false, b, (short)0, c,
                                               false, false);
}

__device__ __forceinline__ float hsum16(float v) {
  v += __shfl_xor(v, 1);
  v += __shfl_xor(v, 2);
  v += __shfl_xor(v, 4);
  v += __shfl_xor(v, 8);
  return v;
}

// ---------------------------------------------------------------------------
// Prepack: P(h,d,n) -> PT(h,n,d)   (k-major so B-fragments are one b64 load)
// ---------------------------------------------------------------------------
__global__ void transpose_p(const float* __restrict__ P, float* __restrict__ PT) {
  int idx = blockIdx.x * blockDim.x + threadIdx.x;   // 0 .. H*DK*D-1
  int h = idx >> 16;          // DK*D = 65536 per head
  int r = idx & 65535;
  int n = r >> 10;
  int k = r & 1023;
  PT[idx] = P[(size_t)h * (DK_ * D_) + (size_t)k * DK_ + n];
}

// P_o(h,d,v) -> WoT(n=d, k=h*64+v): output GEMM B operand, k-major.
__global__ void prepack_wo(const float* __restrict__ Po, float* __restrict__ WoT) {
  int idx = blockIdx.x * blockDim.x + threadIdx.x;   // 0 .. 1024*1024-1
  int d  = idx >> 10;
  int hv = idx & 1023;
  int h = hv >> 6;
  int v = hv & 63;
  WoT[idx] = Po[((size_t)h * D_ + d) * DK_ + v];
}

// ---------------------------------------------------------------------------
// Q/K/V projections. One wave owns a 16x192 tile (12 WMMA accums) of one
// (b,h); K-dim = 1024 in steps of 4. Per k-step: 1 A b64 + 12 B b64 + 12 WMMA.
// Grid: (B*H, M/64), block: 128 (4 waves).
// ---------------------------------------------------------------------------
__global__ __launch_bounds__(128) void qkv_proj(
    const float* __restrict__ x,
    const float* __restrict__ PqT, const float* __restrict__ PkT,
    const float* __restrict__ PvT,
    float* __restrict__ Q, float* __restrict__ K, float* __restrict__ V) {
  const int lane = threadIdx.x & 31;
  const int wave = threadIdx.x >> 5;
  const int ln = lane & 15;   // M (A) / N (B) index within half-wave
  const int hi = lane >> 4;   // selects K pair {0,1} vs {2,3}
  const int bh = blockIdx.x;
  const int b = bh >> 4, h = bh & 15;
  const int m0 = (blockIdx.y * 4 + wave) * 16;

  const float* xa   = x + ((size_t)b * M_ + (m0 + ln)) * D_ + 2 * hi;
  const float* PqTh = PqT + (size_t)h * DK_ * D_;
  const float* PkTh = PkT + (size_t)h * DK_ * D_;
  const float* PvTh = PvT + (size_t)h * DK_ * D_;

  v8f acc[12] = {};

  for (int k0 = 0; k0 < D_; k0 += 4) {
    v2f a = *(const v2f*)(xa + k0);
#pragma unroll
    for (int t = 0; t < 12; ++t) {
      const float* PTh = (t < 4) ? PqTh : (t < 8) ? PkTh : PvTh;
      v2f bf = *(const v2f*)(PTh + (size_t)((t & 3) * 16 + ln) * D_ + k0 + 2 * hi);
      acc[t] = wmma4f32(a, bf, acc[t]);
    }
  }

#pragma unroll
  for (int t = 0; t < 12; ++t) {
    float* Ob = (t < 4) ? Q : (t < 8) ? K : V;
    float* op = Ob + ((size_t)bh * M_ + m0) * DK_ + (t & 3) * 16 + ln;
#pragma unroll
    for (int i = 0; i < 8; ++i) {
      const int row = i + 8 * hi;           // C layout: VGPR i -> rows i / i+8
      op[(size_t)row * DK_] = acc[t][i];
    }
  }
}

// ---------------------------------------------------------------------------
// Causal diag-softmax weights + O = w * V scattered to O[b, m, h*64+v].
// Per-lane online (max, partial-sum); single shuffle merge at the end.
// Grid: (B*H, M/128), block: 256 (8 waves, one 16-row M tile each).
// ---------------------------------------------------------------------------
__global__ __launch_bounds__(256) void attn_diag(
    const float* __restrict__ Q, const float* __restrict__ K,
    const float* __restrict__ V, float* __restrict__ O) {
  const int lane = threadIdx.x & 31;
  const int wave = threadIdx.x >> 5;
  const int ln = lane & 15;
  const int hi = lane >> 4;
  const int bh = blockIdx.x;
  const int b = bh >> 4, h = bh & 15;
  const int mt = blockIdx.y * 8 + wave;
  const int m0 = mt * 16;

  const float* Qb = Q + ((size_t)bh * M_ + m0) * DK_;
  const float* Kb = K + (size_t)bh * M_ * DK_;

  // Preload all 16 A-fragments of the Q tile (16 rows x 64 k).
  v2f aq[16];
#pragma unroll
  for (int s = 0; s < 16; ++s)
    aq[s] = *(const v2f*)(Qb + (size_t)ln * DK_ + 4 * s + 2 * hi);

  const float NEG = -1.0e30f;  // finite sentinel: avoids -inf - -inf = NaN
  float mx[8], ps[8], dl[8];
#pragma unroll
  for (int i = 0; i < 8; ++i) { mx[i] = NEG; ps[i] = 0.0f; dl[i] = 0.0f; }

  for (int nt = 0; nt <= mt; ++nt) {
    const int n0 = nt * 16;
    v8f c = {};
    const float* kb = Kb + (size_t)(n0 + ln) * DK_ + 2 * hi;  // B = K^T
#pragma unroll
    for (int s = 0; s < 16; ++s) {
      v2f bf = *(const v2f*)(kb + 4 * s);
      c = wmma4f32(aq[s], bf, c);
    }
    const bool isdiag = (nt == mt);
#pragma unroll
    for (int i = 0; i < 8; ++i) {
      const int row = i + 8 * hi;
      const float logit = c[i] * 8.0f;              // * sqrt(D_K), faithful
      const bool valid = (!isdiag) || (ln <= row);  // causal mask (diag tile)
      const float lv = valid ? logit : NEG;
      const float nm = fmaxf(mx[i], lv);
      ps[i] = ps[i] * __expf(mx[i] - nm) + (valid ? __expf(logit - nm) : 0.0f);
      mx[i] = nm;
      if (isdiag && (ln == row)) dl[i] = logit;     // l[m,m], one lane per half
    }
  }

#pragma unroll
  for (int i = 0; i < 8; ++i) {
    // Merge the 16 per-lane (max, sum) states of this half.
#pragma unroll
    for (int mask = 1; mask < 16; mask <<= 1) {
      const float om = __shfl_xor(mx[i], mask);
      const float os = __shfl_xor(ps[i], mask);
      const float nm = fmaxf(mx[i], om);
      ps[i] = ps[i] * __expf(mx[i] - nm) + os * __expf(om - nm);
      mx[i] = nm;
    }
    const float dlog = hsum16(dl[i]);               // replicate diag logit
    const float w = __expf(dlog - mx[i]) / ps[i];   // diag softmax weight

    const int row = i + 8 * hi;
    const int mg = m0 + row;
    const float4 vv = ((const float4*)(V + ((size_t)bh * M_ + mg) * DK_))[ln];
    float4 ov = make_float4(vv.x * w, vv.y * w, vv.z * w, vv.w * w);
    ((float4*)(O + ((size_t)b * M_ + mg) * (H_ * DK_) + (size_t)h * DK_))[ln] = ov;
  }
}

// ---------------------------------------------------------------------------
// Y(8192x1024) = O(8192x1024) @ WoT^T. One wave computes 32x64
// (8 accums, 2 A-frags share 4 B-frags): 6 b64 loads per 8 WMMAs.
// Grid: (B*M/32, 4), block: 128 (4 waves -> 256 cols per block).
// ---------------------------------------------------------------------------
__global__ __launch_bounds__(128) void out_proj(
    const float* __restrict__ O, const float* __restrict__ WoT,
    float* __restrict__ Y) {
  const int lane = threadIdx.x & 31;
  const int wave = threadIdx.x >> 5;
  const int ln = lane & 15;
  const int hi = lane >> 4;
  const int m0 = blockIdx.x * 32;                   // row over B*M
  const int n0 = (blockIdx.y * 4 + wave) * 64;

  const float* oa0 = O + (size_t)(m0 + ln) * (H_ * DK_) + 2 * hi;
  const float* oa1 = oa0 + (size_t)16 * (H_ * DK_);
  v8f acc[8] = {};

  for (int k0 = 0; k0 < H_ * DK_; k0 += 4) {
    v2f a0 = *(const v2f*)(oa0 + k0);
    v2f a1 = *(const v2f*)(oa1 + k0);
#pragma unroll
    for (int t = 0; t < 4; ++t) {
      v2f bf = *(const v2f*)(WoT + (size_t)(n0 + t * 16 + ln) * D_ + k0 + 2 * hi);
      acc[t]     = wmma4f32(a0, bf, acc[t]);
      acc[t + 4] = wmma4f32(a1, bf, acc[t + 4]);
    }
  }

#pragma unroll
  for (int t = 0; t < 4; ++t) {
    float* yp = Y + (size_t)m0 * D_ + n0 + t * 16 + ln;
#pragma unroll
    for (int i = 0; i < 8; ++i) {
      const int row = i + 8 * hi;
      yp[(size_t)row * D_]        = acc[t][i];
      yp[(size_t)(row + 16) * D_] = acc[t + 4][i];
    }
  }
}

// ---------------------------------------------------------------------------
extern "C" void kernel_launch(void* const* d_in, const int* in_sizes, int n_in,
                              void* d_out, int out_size, void* d_ws,
                              size_t ws_size, hipStream_t stream) {
  const float* x  = (const float*)d_in[0];
  const float* Pq = (const float*)d_in[1];
  const float* Pk = (const float*)d_in[2];
  const float* Pv = (const float*)d_in[3];
  const float* Po = (const float*)d_in[4];
  float* Y = (float*)d_out;

  const size_t nQ = (size_t)B_ * H_ * M_ * DK_;     // 8.39M floats
  const size_t nP = (size_t)H_ * DK_ * D_;          // 1.05M floats
  float* Q   = (float*)d_ws;
  float* K   = Q + nQ;
  float* V   = K + nQ;
  float* O   = V + nQ;                              // B*M*(H*DK) floats
  float* WoT = O + (size_t)B_ * M_ * H_ * DK_;
  float* PqT = WoT + (size_t)D_ * D_;
  float* PkT = PqT + nP;
  float* PvT = PkT + nP;

  transpose_p<<<dim3(nP / 256), 256, 0, stream>>>(Pq, PqT);
  transpose_p<<<dim3(nP / 256), 256, 0, stream>>>(Pk, PkT);
  transpose_p<<<dim3(nP / 256), 256, 0, stream>>>(Pv, PvT);
  prepack_wo<<<dim3((D_ * D_) / 256), 256, 0, stream>>>(Po, WoT);

  qkv_proj<<<dim3(B_ * H_, M_ / 64), 128, 0, stream>>>(x, PqT, PkT, PvT, Q, K, V);
  attn_diag<<<dim3(B_ * H_, M_ / 128), 256, 0, stream>>>(Q, K, V, O);
  out_proj<<<dim3((B_ * M_) / 32, 4), 128, 0, stream>>>(O, WoT, Y);
}